// DeckGNN_2456721293532
// MI455X (gfx1250) — compile-verified
//
#include <hip/hip_runtime.h>
#include <cstdint>

#define DH 128

typedef __attribute__((ext_vector_type(2))) float v2f;
typedef __attribute__((ext_vector_type(8))) float v8f;

// ---------------------------------------------------------------------------
// Degree kernels
// ---------------------------------------------------------------------------
__global__ __launch_bounds__(256) void zero_i32_kernel(int* __restrict__ p, int n) {
  int i = blockIdx.x * blockDim.x + threadIdx.x;
  if (i < n) p[i] = 0;
}

__global__ __launch_bounds__(256) void count_deg_kernel(const long long* __restrict__ dst,
                                                        int* __restrict__ cnt, long long E) {
  long long e = (long long)blockIdx.x * blockDim.x + threadIdx.x;
  if (e < E) atomicAdd(&cnt[(int)dst[e]], 1);
}

__global__ __launch_bounds__(256) void finalize_deg_kernel(const int* __restrict__ cnt,
                                                           float* __restrict__ isd,
                                                           float* __restrict__ invd, int n) {
  int i = blockIdx.x * blockDim.x + threadIdx.x;
  if (i < n) {
    float deg = 1.0f + (float)cnt[i];
    isd[i]  = rsqrtf(deg);
    invd[i] = 1.0f / deg;
  }
}

__global__ __launch_bounds__(256) void zero_f32x4_kernel(float4* __restrict__ p, long long n4) {
  long long i = (long long)blockIdx.x * blockDim.x + threadIdx.x;
  if (i < n4) p[i] = make_float4(0.f, 0.f, 0.f, 0.f);
}

// ---------------------------------------------------------------------------
// GEMM: H[nrows,128] = X[nrows,128] @ W[128,128] using V_WMMA_F32_16X16X4_F32.
// Block = 256 threads (8 waves). Each block owns a 16-row stripe; wave w owns
// the 16x16 output tile covering columns [16w, 16w+16).
// LDS: full W (64KB) + padded 16x130 X stripe (~8.1KB).
// ---------------------------------------------------------------------------
#define XPITCH 130

__global__ __launch_bounds__(256) void gemm_wmma_f32_kernel(const float* __restrict__ X,
                                                            const float* __restrict__ W,
                                                            float* __restrict__ H, int nrows) {
  __shared__ float sW[DH * DH];       // 64 KB, row-major W[k][n]
  __shared__ float sX[16 * XPITCH];   // padded stripe X[r][k]

  const int tid = threadIdx.x;
  const int row0 = blockIdx.x * 16;

  // Load W: 16384 floats = 4096 float4, 16 per thread.
  for (int i = tid; i < DH * DH / 4; i += 256) {
    ((float4*)sW)[i] = ((const float4*)W)[i];
  }
  // Load X stripe: 16 rows x 128 cols = 512 float4.
  for (int i = tid; i < 512; i += 256) {
    int r  = i >> 5;        // 0..15
    int c4 = i & 31;        // float4 index within row
    int gr = row0 + r;
    float4 v = make_float4(0.f, 0.f, 0.f, 0.f);
    if (gr < nrows) v = ((const float4*)(X + (size_t)gr * DH))[c4];
    float* dstp = &sX[r * XPITCH + c4 * 4];
    dstp[0] = v.x; dstp[1] = v.y; dstp[2] = v.z; dstp[3] = v.w;
  }
  __syncthreads();

  const int wave = tid >> 5;                 // 0..7 -> column tile
  const int lane = tid & 31;
  const int m    = lane & 15;                // row within tile / column within tile
  const int kb   = (lane >> 4) << 1;         // 0 for lanes 0-15, 2 for lanes 16-31
  const int n    = wave * 16 + m;            // global output column

  v8f acc = {0.f, 0.f, 0.f, 0.f, 0.f, 0.f, 0.f, 0.f};

#pragma unroll
  for (int k0 = 0; k0 < DH; k0 += 4) {
    // A fragment (16x4 f32): lane m holds X[m][k0+kb], X[m][k0+kb+1]
    v2f a = *(const v2f*)&sX[m * XPITCH + k0 + kb];
    // B fragment (4x16 f32): lane holds W[k0+kb][n], W[k0+kb+1][n]
    v2f b;
    b.x = sW[(k0 + kb) * DH + n];
    b.y = sW[(k0 + kb + 1) * DH + n];
    acc = __builtin_amdgcn_wmma_f32_16x16x4_f32(false, a, false, b,
                                                (short)0, acc, false, false);
  }

  // C/D layout: VGPR r holds row (r + 0) for lanes 0-15, row (r + 8) for 16-31.
  const int rbase = row0 + ((lane >> 4) << 3);
#pragma unroll
  for (int r = 0; r < 8; ++r) {
    int gr = rbase + r;
    if (gr < nrows) H[(size_t)gr * DH + n] = acc[r];
  }
}

// ---------------------------------------------------------------------------
// Edge scatter: one wave per edge, 4 floats per lane, native f32 atomics.
// C[dst] += H[src] * (isd[src]*isd[dst])
// ---------------------------------------------------------------------------
__global__ __launch_bounds__(256) void edge_scatter_kernel(const float* __restrict__ H,
                                                           const long long* __restrict__ src,
                                                           const long long* __restrict__ dst,
                                                           const float* __restrict__ isd,
                                                           float* __restrict__ C, long long E) {
  long long t = (long long)blockIdx.x * blockDim.x + threadIdx.x;
  long long e = t >> 5;
  if (e >= E) return;
  int lane = (int)(t & 31);
  long long s = src[e];
  long long d = dst[e];
  float ns = isd[s] * isd[d];
  float4 v = ((const float4*)(H + s * DH))[lane];
  float* cp = C + d * DH + lane * 4;
  unsafeAtomicAdd(cp + 0, v.x * ns);
  unsafeAtomicAdd(cp + 1, v.y * ns);
  unsafeAtomicAdd(cp + 2, v.z * ns);
  unsafeAtomicAdd(cp + 3, v.w * ns);
}

// ---------------------------------------------------------------------------
// Combine: H = relu(C + H * invd[node] + b[col])   (in-place on H)
// ---------------------------------------------------------------------------
__global__ __launch_bounds__(256) void combine_kernel(const float* __restrict__ C,
                                                      const float* __restrict__ invd,
                                                      const float* __restrict__ b,
                                                      float* __restrict__ H, long long total) {
  long long i = (long long)blockIdx.x * blockDim.x + threadIdx.x;
  if (i >= total) return;
  int col = (int)(i & (DH - 1));
  long long node = i >> 7;
  float v = C[i] + H[i] * invd[node] + b[col];
  H[i] = fmaxf(v, 0.f);
}

// ---------------------------------------------------------------------------
// Output head: out[i] = sigmoid(dot(H[i,:], Wout) + bout)
// ---------------------------------------------------------------------------
__global__ __launch_bounds__(256) void head_kernel(const float* __restrict__ H,
                                                   const float* __restrict__ Wout,
                                                   const float* __restrict__ bout,
                                                   float* __restrict__ out, int n) {
  __shared__ float sw[DH];
  if (threadIdx.x < DH) sw[threadIdx.x] = Wout[threadIdx.x];
  __syncthreads();
  int i = blockIdx.x * blockDim.x + threadIdx.x;
  if (i >= n) return;
  const float4* hp = (const float4*)(H + (size_t)i * DH);
  float acc = 0.f;
#pragma unroll
  for (int j = 0; j < DH / 4; ++j) {
    float4 v = hp[j];
    acc += v.x * sw[4 * j] + v.y * sw[4 * j + 1] + v.z * sw[4 * j + 2] + v.w * sw[4 * j + 3];
  }
  acc += bout[0];
  out[i] = 1.0f / (1.0f + expf(-acc));
}

// ---------------------------------------------------------------------------
// Launch
// ---------------------------------------------------------------------------
extern "C" void kernel_launch(void* const* d_in, const int* in_sizes, int n_in,
                              void* d_out, int out_size, void* d_ws, size_t ws_size,
                              hipStream_t stream) {
  const float*      x    = (const float*)d_in[0];
  const long long*  ei   = (const long long*)d_in[1];   // [2, E] int64
  const float*      W1   = (const float*)d_in[2];
  const float*      b1   = (const float*)d_in[3];
  const float*      W2   = (const float*)d_in[4];
  const float*      b2   = (const float*)d_in[5];
  const float*      Wout = (const float*)d_in[6];
  const float*      bout = (const float*)d_in[7];
  float*            out  = (float*)d_out;

  const int       N = in_sizes[0] / DH;
  const long long E = in_sizes[1] / 2;
  const long long* src = ei;
  const long long* dst = ei + E;

  // Workspace layout
  char*  ws   = (char*)d_ws;
  size_t feat = (size_t)N * DH * sizeof(float);
  float* A      = (float*)(ws);                 // feature buffer 1
  float* B      = (float*)(ws + feat);          // feature buffer 2
  float* C      = (float*)(ws + 2 * feat);      // scatter accumulator
  int*   degcnt = (int*)  (ws + 3 * feat);
  float* isd    = (float*)(ws + 3 * feat + (size_t)N * 4);
  float* invd   = (float*)(ws + 3 * feat + (size_t)N * 8);

  const long long totalF = (long long)N * DH;
  const long long n4     = totalF / 4;

  dim3 blk(256);
  unsigned gN      = (unsigned)((N + 255) / 256);
  unsigned gE      = (unsigned)((E + 255) / 256);
  unsigned gE32    = (unsigned)((E * 32 + 255) / 256);
  unsigned gF      = (unsigned)((totalF + 255) / 256);
  unsigned gF4     = (unsigned)((n4 + 255) / 256);
  unsigned gRows   = (unsigned)((N + 15) / 16);

  // Degrees (with self-loop): deg = 1 + incoming count
  zero_i32_kernel<<<gN, blk, 0, stream>>>(degcnt, N);
  count_deg_kernel<<<gE, blk, 0, stream>>>(dst, degcnt, E);
  finalize_deg_kernel<<<gN, blk, 0, stream>>>(degcnt, isd, invd, N);

  // ---- Layer 1 ----
  gemm_wmma_f32_kernel<<<gRows, blk, 0, stream>>>(x, W1, A, N);
  zero_f32x4_kernel<<<gF4, blk, 0, stream>>>((float4*)C, n4);
  edge_scatter_kernel<<<gE32, blk, 0, stream>>>(A, src, dst, isd, C, E);
  combine_kernel<<<gF, blk, 0, stream>>>(C, invd, b1, A, totalF);

  // ---- Layer 2 ----
  gemm_wmma_f32_kernel<<<gRows, blk, 0, stream>>>(A, W2, B, N);
  zero_f32x4_kernel<<<gF4, blk, 0, stream>>>((float4*)C, n4);
  edge_scatter_kernel<<<gE32, blk, 0, stream>>>(B, src, dst, isd, C, E);
  combine_kernel<<<gF, blk, 0, stream>>>(C, invd, b2, B, totalF);

  // ---- Output head ----
  head_kernel<<<gN, blk, 0, stream>>>(B, Wout, bout, out, N);
}